// GCN__30339648979444
// MI455X (gfx1250) — compile-verified
//
#include <hip/hip_runtime.h>

typedef __attribute__((ext_vector_type(2))) float v2f;
typedef __attribute__((ext_vector_type(8))) float v8f;

#define IN_FT 128
#define OUT_FT 32

// ---------------------------------------------------------------------------
// Dense projection: support[n, 32] = seq[n, 128] @ W[128, 32]
// One 16-row strip per block; 2 waves, each owns one 16-col N-tile.
// v_wmma_f32_16x16x4_f32, K stepped by 4 (32 WMMAs per tile).
//
// A (16x4 f32) layout: lanes 0-15 -> {K+0 in v0, K+1 in v1},
//                      lanes 16-31 -> {K+2 in v0, K+3 in v1}  => float2 load.
// B (4x16 f32) layout mirrors the K split across lane halves.
// D (16x16 f32): VGPR r, lanes 0-15 -> M=r ; lanes 16-31 -> M=r+8.
// ---------------------------------------------------------------------------
__global__ __launch_bounds__(64) void gcn_gemm_wmma(
    const float* __restrict__ seq, const float* __restrict__ W,
    float* __restrict__ support, int n_nodes) {
  const int row0 = blockIdx.x << 4;          // 16 rows per block
  const int lane = threadIdx.x & 31;
  const int wave = threadIdx.x >> 5;         // which 16-col N tile
  const int n0   = wave << 4;
  const int lh   = lane & 15;
  const int koff = (lane >> 4) << 1;         // 0 for lanes 0-15, 2 for 16-31

  const float* arow = seq + (size_t)(row0 + lh) * IN_FT;
  const float* bcol = W + n0 + lh;

  v8f acc = (v8f)0.0f;
#pragma unroll 4
  for (int k = 0; k < IN_FT; k += 4) {
    v2f a = *(const v2f*)(arow + k + koff);          // contiguous 8B, aligned
    v2f b;
    b.x = bcol[(size_t)(k + koff) * OUT_FT];
    b.y = bcol[(size_t)(k + koff + 1) * OUT_FT];
    acc = __builtin_amdgcn_wmma_f32_16x16x4_f32(
        /*neg_a=*/false, a, /*neg_b=*/false, b,
        /*c_mod=*/(short)0, acc, /*reuse_a=*/false, /*reuse_b=*/false);
  }

  const int mbase = row0 + ((lane >> 4) << 3);       // +8 rows for hi half
  float* crow = support + (size_t)mbase * OUT_FT + n0 + lh;
#pragma unroll
  for (int r = 0; r < 8; ++r)
    crow[(size_t)r * OUT_FT] = acc[r];
}

// ---------------------------------------------------------------------------
// Zero-init of the accumulation buffer (must run every call: atomics below).
// ---------------------------------------------------------------------------
__global__ __launch_bounds__(256) void gcn_zero4(float4* __restrict__ p, int n4) {
  int i = blockIdx.x * blockDim.x + threadIdx.x;
  if (i < n4) p[i] = make_float4(0.f, 0.f, 0.f, 0.f);
}

// ---------------------------------------------------------------------------
// COO SpMM scatter: out[row_e, :] += val_e * support[col_e, :]
// One wave per edge, lane = feature. 128B coalesced gather from L2-resident
// support, 128B coalesced global_atomic_add_f32 scatter.
// ---------------------------------------------------------------------------
__global__ __launch_bounds__(256) void gcn_spmm_atomic(
    const float* __restrict__ support, const int* __restrict__ erow,
    const int* __restrict__ ecol, const float* __restrict__ eval,
    float* __restrict__ out, int n_edges) {
  const int lane = threadIdx.x & 31;
  const int e = blockIdx.x * (blockDim.x >> 5) + (threadIdx.x >> 5);
  if (e >= n_edges) return;
  const int r = erow[e];
  const int c = ecol[e];
  const float v = eval[e];
  const float m = v * support[(size_t)c * OUT_FT + lane];
  atomicAdd(out + (size_t)r * OUT_FT + lane, m);
}

// ---------------------------------------------------------------------------
// ReLU (separate pass; atomics prevent fusion).
// ---------------------------------------------------------------------------
__global__ __launch_bounds__(256) void gcn_relu4(float4* __restrict__ p, int n4) {
  int i = blockIdx.x * blockDim.x + threadIdx.x;
  if (i < n4) {
    float4 x = p[i];
    x.x = fmaxf(x.x, 0.f);
    x.y = fmaxf(x.y, 0.f);
    x.z = fmaxf(x.z, 0.f);
    x.w = fmaxf(x.w, 0.f);
    p[i] = x;
  }
}

extern "C" void kernel_launch(void* const* d_in, const int* in_sizes, int n_in,
                              void* d_out, int out_size, void* d_ws, size_t ws_size,
                              hipStream_t stream) {
  const float* seq  = (const float*)d_in[0];   // [N, 128] f32
  const float* W    = (const float*)d_in[1];   // [128, 32] f32
  const int*   erow = (const int*)d_in[2];     // [E] int
  const int*   ecol = (const int*)d_in[3];     // [E] int
  const float* eval = (const float*)d_in[4];   // [E] f32
  float* out = (float*)d_out;                  // [N, 32] f32

  const int n_nodes = in_sizes[0] / IN_FT;
  const int n_edges = in_sizes[2];

  float* support = (float*)d_ws;               // [N, 32] f32 scratch (12.8 MB)

  // 1) zero the accumulator
  const int n4 = out_size >> 2;
  gcn_zero4<<<(n4 + 255) / 256, 256, 0, stream>>>((float4*)out, n4);

  // 2) dense projection via fp32 WMMA
  gcn_gemm_wmma<<<(n_nodes + 15) / 16, 64, 0, stream>>>(seq, W, support, n_nodes);

  // 3) edge scatter with fp32 atomics (one wave per edge)
  const int edges_per_block = 256 / 32;
  gcn_spmm_atomic<<<(n_edges + edges_per_block - 1) / edges_per_block, 256, 0,
                    stream>>>(support, erow, ecol, eval, out, n_edges);

  // 4) ReLU
  gcn_relu4<<<(n4 + 255) / 256, 256, 0, stream>>>((float4*)out, n4);
}